// GrapHiC_53541062312585
// MI455X (gfx1250) — compile-verified
//
#include <hip/hip_runtime.h>
#include <math.h>

#define HH   4
#define CCH  32
#define HC   128
#define BB   8
#define NPER 2048
#define NN   (BB * NPER)      // 16384 nodes
#define EDG  1048576          // edges
#define EPSV 1e-5f

#define NHC  (NN * HC)        // 2,097,152
#define NH   (NN * HH)        // 65,536
#define NC   (NN * CCH)       // 524,288
#define EH   (EDG * HH)       // 4,194,304

typedef __attribute__((ext_vector_type(2))) float v2f;
typedef __attribute__((ext_vector_type(8))) float v8f;

__device__ __forceinline__ void atomicMaxF(float* addr, float val) {
    // ordered-int trick: correct float max for any sign mix
    if (val >= 0.0f) atomicMax((int*)addr, __float_as_int(val));
    else             atomicMin((unsigned int*)addr, __float_as_uint(val));
}

__device__ __forceinline__ float fast_sigmoid(float x) {
    // v_exp_f32 + v_rcp_f32 (avoids IEEE divide expansion)
    return __builtin_amdgcn_rcpf(1.0f + __expf(-x));
}

// ---------------- init scratch (acc, denom, amax) ----------------
__global__ void k_init(float* acc, float* denom, float* amax) {
    int i = blockIdx.x * blockDim.x + threadIdx.x;
    if (i < NHC) acc[i] = 0.0f;
    if (i < NH) { denom[i] = 0.0f; amax[i] = -INFINITY; }
}

// ---------------- node projections: q,k,v,x_r ----------------
__global__ void k_proj(const float* __restrict__ x,
                       const float* __restrict__ Wq, const float* __restrict__ bq,
                       const float* __restrict__ Wk, const float* __restrict__ bk,
                       const float* __restrict__ Wv, const float* __restrict__ bv,
                       const float* __restrict__ Ws, const float* __restrict__ bs,
                       float* __restrict__ q, float* __restrict__ k,
                       float* __restrict__ v, float* __restrict__ xr) {
    int idx = blockIdx.x * blockDim.x + threadIdx.x;
    if (idx >= NHC) return;
    int n = idx >> 7, j = idx & 127;
    float x0 = x[n*4+0], x1 = x[n*4+1], x2 = x[n*4+2], x3 = x[n*4+3];
    q[idx]  = bq[j] + x0*Wq[j] + x1*Wq[HC+j] + x2*Wq[2*HC+j] + x3*Wq[3*HC+j];
    k[idx]  = bk[j] + x0*Wk[j] + x1*Wk[HC+j] + x2*Wk[2*HC+j] + x3*Wk[3*HC+j];
    v[idx]  = bv[j] + x0*Wv[j] + x1*Wv[HC+j] + x2*Wv[2*HC+j] + x3*Wv[3*HC+j];
    xr[idx] = bs[j] + x0*Ws[j] + x1*Ws[HC+j] + x2*Ws[2*HC+j] + x3*Ws[3*HC+j];
}

// ---------------- edge pass A: alpha logits + segment max ----------------
__global__ void k_edgeA(const int* __restrict__ ei, const float* __restrict__ ea,
                        const float* __restrict__ q, const float* __restrict__ kk,
                        const float* __restrict__ We, const float* __restrict__ be,
                        float* __restrict__ alpha, float* __restrict__ amax) {
    int t = blockIdx.x * blockDim.x + threadIdx.x;
    if (t >= EH) return;
    int e = t >> 2, h = t & 3;
    int s = ei[e], d = ei[EDG + e];
    float a = ea[e];
    const float4* qd = (const float4*)(q  + (size_t)d*HC + h*CCH);
    const float4* ks = (const float4*)(kk + (size_t)s*HC + h*CCH);
    const float4* w  = (const float4*)(We + h*CCH);
    const float4* bb = (const float4*)(be + h*CCH);
    float dot = 0.0f;
#pragma unroll
    for (int i = 0; i < 8; ++i) {
        float4 qv = qd[i], kv = ks[i], wv = w[i], bv = bb[i];
        dot += qv.x*(kv.x + a*wv.x + bv.x);
        dot += qv.y*(kv.y + a*wv.y + bv.y);
        dot += qv.z*(kv.z + a*wv.z + bv.z);
        dot += qv.w*(kv.w + a*wv.w + bv.w);
    }
    dot *= 0.17677669529663687f;  // 1/sqrt(32)
    alpha[t] = dot;
    atomicMaxF(&amax[d*HH + h], dot);
}

// ---------------- edge pass B: exp, denom, weighted-V scatter ----------------
__global__ void k_edgeB(const int* __restrict__ ei, const float* __restrict__ ea,
                        const float* __restrict__ v, const float* __restrict__ We,
                        const float* __restrict__ be,
                        const float* __restrict__ alpha, const float* __restrict__ amax,
                        float* __restrict__ denom, float* __restrict__ acc) {
    int t = blockIdx.x * blockDim.x + threadIdx.x;
    if (t >= EH) return;
    int e = t >> 2, h = t & 3;
    int s = ei[e], d = ei[EDG + e];
    float a = ea[e];
    float p = __expf(alpha[t] - amax[d*HH + h]);
    atomicAdd(&denom[d*HH + h], p);
    const float* vs = v + (size_t)s*HC + h*CCH;
    const float* w  = We + h*CCH;
    const float* bb = be + h*CCH;
    float* ac = acc + (size_t)d*HC + h*CCH;
#pragma unroll
    for (int c = 0; c < CCH; ++c)
        atomicAdd(&ac[c], p * (vs[c] + a*w[c] + bb[c]));
}

// ---------------- node epilogue: softmax norm, beta gate, Linear+ReLU ----------------
__global__ void __launch_bounds__(HC) k_node(const float* __restrict__ acc,
                                             const float* __restrict__ denom,
                                             const float* __restrict__ xr,
                                             const float* __restrict__ Wbeta,
                                             const float* __restrict__ Wl,
                                             const float* __restrict__ bl,
                                             float* __restrict__ hout) {
    __shared__ float red[HC];
    __shared__ float ob[HC];
    int n = blockIdx.x, j = threadIdx.x;
    float den = denom[n*HH + (j >> 5)];
    den = den > 0.0f ? den : 1.0f;
    float o = acc[(size_t)n*HC + j] / den;
    float r = xr[(size_t)n*HC + j];
    red[j] = o*Wbeta[j] + r*Wbeta[HC + j] + (o - r)*Wbeta[2*HC + j];
    __syncthreads();
    for (int st = 64; st > 0; st >>= 1) {
        if (j < st) red[j] += red[j + st];
        __syncthreads();
    }
    float beta = fast_sigmoid(red[0]);
    ob[j] = beta*r + (1.0f - beta)*o;
    __syncthreads();
    if (j < CCH) {
        float s = bl[j];
#pragma unroll 8
        for (int i = 0; i < HC; ++i) s += ob[i] * Wl[i*CCH + j];
        hout[(size_t)n*CCH + j] = fmaxf(s, 0.0f);
    }
}

// ---------------- GraphNorm: mean per (graph, channel) ----------------
__global__ void k_mean(const float* __restrict__ h, float* __restrict__ mean) {
    __shared__ float red[256];
    int b = blockIdx.x >> 5, c = blockIdx.x & 31;
    float s = 0.0f;
    for (int i = threadIdx.x; i < NPER; i += 256)
        s += h[((size_t)(b*NPER + i))*CCH + c];
    red[threadIdx.x] = s; __syncthreads();
    for (int st = 128; st > 0; st >>= 1) {
        if (threadIdx.x < st) red[threadIdx.x] += red[threadIdx.x + st];
        __syncthreads();
    }
    if (threadIdx.x == 0) mean[blockIdx.x] = red[0] * (1.0f / NPER);
}

// ---------------- GraphNorm: inv-std per (graph, channel) ----------------
__global__ void k_var(const float* __restrict__ h, const float* __restrict__ mean,
                      const float* __restrict__ gn_ms, float* __restrict__ istd) {
    __shared__ float red[256];
    int b = blockIdx.x >> 5, c = blockIdx.x & 31;
    float m = gn_ms[c] * mean[blockIdx.x];
    float s = 0.0f;
    for (int i = threadIdx.x; i < NPER; i += 256) {
        float d = h[((size_t)(b*NPER + i))*CCH + c] - m;
        s += d * d;
    }
    red[threadIdx.x] = s; __syncthreads();
    for (int st = 128; st > 0; st >>= 1) {
        if (threadIdx.x < st) red[threadIdx.x] += red[threadIdx.x + st];
        __syncthreads();
    }
    if (threadIdx.x == 0) istd[blockIdx.x] = rsqrtf(red[0] * (1.0f / NPER) + EPSV);
}

// ---------------- GraphNorm: normalize -> Z ----------------
__global__ void k_zn(const float* __restrict__ h, const float* __restrict__ mean,
                     const float* __restrict__ istd, const float* __restrict__ gn_ms,
                     const float* __restrict__ gn_w, const float* __restrict__ gn_b,
                     float* __restrict__ Z) {
    int idx = blockIdx.x * blockDim.x + threadIdx.x;
    if (idx >= NC) return;
    int c = idx & 31, n = idx >> 5, b = n >> 11;
    float m = gn_ms[c] * mean[b*CCH + c];
    Z[idx] = (h[idx] - m) * istd[b*CCH + c] * gn_w[c] + gn_b[c];
}

// ---------------- decoder: sigmoid(Z Z^T) via V_WMMA_F32_16X16X4_F32 ----------------
// one wave32 per 16x64 output strip: A fragments loaded once, reused for 4
// column tiles (32 chained K=4 WMMAs per wave). Full f32 precision; the stage
// is HBM-store bound (134 MB output) so f32 WMMA costs nothing vs f16.
__global__ void __launch_bounds__(256) k_dec(const float* __restrict__ Z,
                                             float* __restrict__ out) {
    int wave = threadIdx.x >> 5;
    int lane = threadIdx.x & 31;
    int lm = lane & 15, hi = lane >> 4;
    int strip = blockIdx.x * 8 + wave;        // 8 * 128 * 32 = 32768 strips
    int b   = strip >> 12;                    // 4096 strips per graph
    int t   = strip & 4095;
    int mt  = t >> 5;                         // 128 row tiles
    int nt4 = t & 31;                         // 32 col strips (64 cols each)
    const float* Zb = Z + (size_t)b * NPER * CCH;
    // A (16x4): lane=M%16; VGPR0 holds K={0,2}, VGPR1 K={1,3} by lane half
    const float* ar = Zb + (size_t)(mt*16 + lm) * CCH + 2*hi;
    v2f a[8];
#pragma unroll
    for (int s = 0; s < 8; ++s) { a[s][0] = ar[4*s]; a[s][1] = ar[4*s + 1]; }
    float* ob = out + (size_t)b * NPER * NPER;
#pragma unroll
    for (int u = 0; u < 4; ++u) {
        int nt = nt4*4 + u;
        // B (4x16): same packing with N in place of M (row-major Z)
        const float* br = Zb + (size_t)(nt*16 + lm) * CCH + 2*hi;
        v8f c = {};
#pragma unroll
        for (int s = 0; s < 8; ++s) {
            v2f bv; bv[0] = br[4*s]; bv[1] = br[4*s + 1];
            c = __builtin_amdgcn_wmma_f32_16x16x4_f32(
                    /*neg_a=*/false, a[s], /*neg_b=*/false, bv,
                    /*c_mod=*/(short)0, c, /*reuse_a=*/false, /*reuse_b=*/false);
        }
        int col = nt*16 + lm;
#pragma unroll
        for (int j = 0; j < 8; ++j) {
            int row = mt*16 + j + 8*hi;       // C/D layout: VGPR j -> M = j + 8*hi
            ob[(size_t)row * NPER + col] = fast_sigmoid(c[j]);
        }
    }
}

extern "C" void kernel_launch(void* const* d_in, const int* in_sizes, int n_in,
                              void* d_out, int out_size, void* d_ws, size_t ws_size,
                              hipStream_t stream) {
    const float* x   = (const float*)d_in[0];
    const int*   ei  = (const int*)  d_in[1];
    const float* ea  = (const float*)d_in[2];
    // d_in[3] batch_index implicit (node n -> graph n/2048)
    const float* Wq  = (const float*)d_in[4],  *bq = (const float*)d_in[5];
    const float* Wk  = (const float*)d_in[6],  *bk = (const float*)d_in[7];
    const float* Wv  = (const float*)d_in[8],  *bv = (const float*)d_in[9];
    const float* We  = (const float*)d_in[10], *be = (const float*)d_in[11];
    const float* Ws  = (const float*)d_in[12], *bs = (const float*)d_in[13];
    const float* Wbeta = (const float*)d_in[14];
    const float* Wl  = (const float*)d_in[15], *bl = (const float*)d_in[16];
    const float* gnw = (const float*)d_in[17], *gnb = (const float*)d_in[18];
    const float* gnms= (const float*)d_in[19];

    float* ws    = (float*)d_ws;
    float* q     = ws;                 // NHC
    float* k     = q     + NHC;        // NHC
    float* v     = k     + NHC;        // NHC
    float* xr    = v     + NHC;        // NHC
    float* acc   = xr    + NHC;        // NHC
    float* alpha = acc   + NHC;        // EH
    float* amax  = alpha + EH;         // NH
    float* denom = amax  + NH;         // NH
    float* hbuf  = denom + NH;         // NC
    float* Zbuf  = hbuf  + NC;         // NC
    float* meanb = Zbuf  + NC;         // 256
    float* istdb = meanb + 256;        // 256   -> total ~63.4 MB

    k_init <<<(NHC + 255)/256, 256, 0, stream>>>(acc, denom, amax);
    k_proj <<<(NHC + 255)/256, 256, 0, stream>>>(x, Wq,bq, Wk,bk, Wv,bv, Ws,bs, q,k,v,xr);
    k_edgeA<<<EH/256, 256, 0, stream>>>(ei, ea, q, k, We, be, alpha, amax);
    k_edgeB<<<EH/256, 256, 0, stream>>>(ei, ea, v, We, be, alpha, amax, denom, acc);
    k_node <<<NN, HC, 0, stream>>>(acc, denom, xr, Wbeta, Wl, bl, hbuf);
    k_mean <<<BB*CCH, 256, 0, stream>>>(hbuf, meanb);
    k_var  <<<BB*CCH, 256, 0, stream>>>(hbuf, meanb, gnms, istdb);
    k_zn   <<<(NC + 255)/256, 256, 0, stream>>>(hbuf, meanb, istdb, gnms, gnw, gnb, Zbuf);
    k_dec  <<<32768/8, 256, 0, stream>>>(Zbuf, (float*)d_out);
}